// SetAbstraction_8452495638616
// MI455X (gfx1250) — compile-verified
//
#include <hip/hip_runtime.h>
#include <hip/hip_bf16.h>

// ---------------------------------------------------------------------------
// PointNet++ SetAbstraction for MI455X (gfx1250, wave32, WMMA).
//   1) fps_kernel        : farthest point sampling (1 block per batch)
//   2) ball_group_kernel : ball query + build X (19ch x M) grouped matrix
//   3) gemm_wmma<K>      : Y = W @ X via V_WMMA_F32_16X16X4_F32, 4 tiles/wave
//   4) stats_partial/reduce : deterministic per-channel sum/sumsq for BN
//   5) bnrelu            : in-place BN(training stats)+ReLU
//   6) maxpool_bn        : BN3+ReLU+max over nsample -> output features
// Workspace requirement: ~592 MB.
// ---------------------------------------------------------------------------

#define B_      8
#define N_      8192
#define CF      16          // input feature channels
#define S_      2048        // NPOINT
#define NS      64          // NSAMPLE
#define M_      (B_ * S_ * NS)   // 1048576 = 2^20 positions
#define EPSBN   1e-5f

typedef __attribute__((ext_vector_type(2))) float v2f;
typedef __attribute__((ext_vector_type(8))) float v8f;

// ---------------------------------------------------------------- FPS ------
__global__ __launch_bounds__(1024)
void fps_kernel(const float* __restrict__ xyz, int* __restrict__ fps_idx,
                float* __restrict__ new_xyz) {
    const int b = blockIdx.x;
    const int t = threadIdx.x;                 // 0..1023
    const float* px = xyz + (size_t)b * N_ * 3;

    float X[8], Y[8], Z[8], D[8];
#pragma unroll
    for (int k = 0; k < 8; ++k) {
        int p = k * 1024 + t;
        X[k] = px[p * 3 + 0];
        Y[k] = px[p * 3 + 1];
        Z[k] = px[p * 3 + 2];
        D[k] = 1e10f;
    }

    __shared__ float sval[32];
    __shared__ int   sidx[32];
    __shared__ int   sbest;

    if (t == 0) {
        fps_idx[b * S_] = 0;
        new_xyz[((size_t)b * S_) * 3 + 0] = px[0];
        new_xyz[((size_t)b * S_) * 3 + 1] = px[1];
        new_xyz[((size_t)b * S_) * 3 + 2] = px[2];
    }

    int last = 0;
    for (int it = 1; it < S_; ++it) {
        const float lx = px[last * 3 + 0];
        const float ly = px[last * 3 + 1];
        const float lz = px[last * 3 + 2];

        float bv = -1.0f;
        int   bi = 0x7fffffff;
#pragma unroll
        for (int k = 0; k < 8; ++k) {
            float dx = X[k] - lx, dy = Y[k] - ly, dz = Z[k] - lz;
            float d  = dx * dx + dy * dy + dz * dz;
            d = fminf(D[k], d);
            D[k] = d;
            int p = k * 1024 + t;
            if (d > bv || (d == bv && p < bi)) { bv = d; bi = p; }
        }
        // wave32 reduction (argmax, ties -> lowest index, matches jnp.argmax)
#pragma unroll
        for (int m = 16; m >= 1; m >>= 1) {
            float ov = __shfl_xor(bv, m, 32);
            int   oi = __shfl_xor(bi, m, 32);
            if (ov > bv || (ov == bv && oi < bi)) { bv = ov; bi = oi; }
        }
        if ((t & 31) == 0) { sval[t >> 5] = bv; sidx[t >> 5] = bi; }
        __syncthreads();
        if (t < 32) {
            bv = sval[t]; bi = sidx[t];
#pragma unroll
            for (int m = 16; m >= 1; m >>= 1) {
                float ov = __shfl_xor(bv, m, 32);
                int   oi = __shfl_xor(bi, m, 32);
                if (ov > bv || (ov == bv && oi < bi)) { bv = ov; bi = oi; }
            }
            if (t == 0) {
                sbest = bi;
                fps_idx[b * S_ + it] = bi;
                new_xyz[((size_t)b * S_ + it) * 3 + 0] = px[bi * 3 + 0];
                new_xyz[((size_t)b * S_ + it) * 3 + 1] = px[bi * 3 + 1];
                new_xyz[((size_t)b * S_ + it) * 3 + 2] = px[bi * 3 + 2];
            }
        }
        __syncthreads();
        last = sbest;
    }
}

// ------------------------------------------------- ball query + grouping ---
__global__ __launch_bounds__(256)
void ball_group_kernel(const float* __restrict__ xyz,
                       const float* __restrict__ feats,
                       const float* __restrict__ new_xyz,
                       float* __restrict__ Xout) {
    const int lane = threadIdx.x & 31;
    const int wib  = threadIdx.x >> 5;
    const int cid  = blockIdx.x * 8 + wib;     // centroid id, 0..16383
    const int b    = cid >> 11;                // / 2048
    const float cx = new_xyz[(size_t)cid * 3 + 0];
    const float cy = new_xyz[(size_t)cid * 3 + 1];
    const float cz = new_xyz[(size_t)cid * 3 + 2];
    const float* px = xyz + (size_t)b * N_ * 3;
    const float r2 = 0.01f;

    __shared__ volatile int lists[8][NS];
    volatile int* list = lists[wib];

    int cnt = 0;
    for (int c0 = 0; c0 < N_ && cnt < NS; c0 += 32) {
        int p = c0 + lane;
        float dx = px[p * 3 + 0] - cx;
        float dy = px[p * 3 + 1] - cy;
        float dz = px[p * 3 + 2] - cz;
        float d2 = dx * dx + dy * dy + dz * dz;
        bool hit = d2 < r2;
        unsigned mask = (unsigned)__ballot(hit);
        if (hit) {
            int pos = cnt + __popc(mask & ((1u << lane) - 1u));
            if (pos < NS) list[pos] = p;
        }
        cnt += __popc(mask);
    }
    if (cnt > NS) cnt = NS;
    int first = list[0];   // centroid is always within radius -> cnt >= 1
    for (int j = cnt + lane; j < NS; j += 32) list[j] = first;

    // Emit grouped matrix: rows 0..2 = xyz - center, rows 3..18 = features,
    // row 19 = zero K-padding. Channel-major, M contiguous.
    const float* pf = feats + (size_t)b * CF * N_;
    for (int j = lane; j < NS; j += 32) {
        int idx = list[j];
        size_t m = (size_t)cid * NS + j;
        Xout[0 * (size_t)M_ + m] = px[idx * 3 + 0] - cx;
        Xout[1 * (size_t)M_ + m] = px[idx * 3 + 1] - cy;
        Xout[2 * (size_t)M_ + m] = px[idx * 3 + 2] - cz;
#pragma unroll
        for (int c = 0; c < CF; ++c)
            Xout[(size_t)(3 + c) * M_ + m] = pf[(size_t)c * N_ + idx];
        Xout[(size_t)19 * M_ + m] = 0.0f;
    }
}

// ----------------------------------------------------- WMMA fp32 GEMM ------
// Y(Cout x M) = W(Cout x KACT, row-major) @ Xin(KPAD x M, channel-major).
// One wave -> 16 x (16*NT) output tile via V_WMMA_F32_16X16X4_F32.
// K is compile-time: loops fully unroll, no divergent guarded loads.
template<int KACT, int KPAD, int NT>
__global__ __launch_bounds__(256)
void gemm_wmma(const float* __restrict__ Wm, const float* __restrict__ Xin,
               float* __restrict__ Yout) {
    const int lane = threadIdx.x & 31;
    const int n16  = lane & 15;       // A: row M ; B/D: column N
    const int kh   = lane >> 4;       // K-half select
    const int n0   = (blockIdx.x * 8 + (threadIdx.x >> 5)) * (16 * NT);
    const int r0   = blockIdx.y * 16;

    // Preload all A (weight) fragments once; unconditional clamped loads +
    // select-to-zero for the pad lanes (no exec-mask branches).
    v2f A[KPAD / 4];
#pragma unroll
    for (int s = 0; s < KPAD / 4; ++s) {
        const int ka = s * 4 + 2 * kh;
        int i0 = ka     < KACT ? ka     : KACT - 1;
        int i1 = ka + 1 < KACT ? ka + 1 : KACT - 1;
        float a0 = Wm[(r0 + n16) * KACT + i0];
        float a1 = Wm[(r0 + n16) * KACT + i1];
        a0 = (ka     < KACT) ? a0 : 0.0f;
        a1 = (ka + 1 < KACT) ? a1 : 0.0f;
        A[s] = v2f{a0, a1};
    }

    v8f acc[NT];
#pragma unroll
    for (int t = 0; t < NT; ++t) acc[t] = v8f{};

#pragma unroll
    for (int s = 0; s < KPAD / 4; ++s) {
        const int ka = s * 4 + 2 * kh;
        const float* rp0 = Xin + (size_t)ka * M_ + n0 + n16;
        const float* rp1 = rp0 + M_;
#pragma unroll
        for (int t = 0; t < NT; ++t) {
            v2f Bv = v2f{rp0[16 * t], rp1[16 * t]};
            acc[t] = __builtin_amdgcn_wmma_f32_16x16x4_f32(
                         false, A[s], false, Bv, (short)0, acc[t],
                         false, false);
        }
    }

    // D: VGPR r -> row r (lanes 0-15) / row r+8 (lanes 16-31), col = n16
#pragma unroll
    for (int r = 0; r < 8; ++r) {
        const int mm = r0 + r + 8 * kh;
        float* yp = Yout + (size_t)mm * M_ + n0 + n16;
#pragma unroll
        for (int t = 0; t < NT; ++t)
            yp[16 * t] = acc[t][r];
    }
}

// -------------------------------- deterministic BN statistics (sum,sumsq) --
__global__ __launch_bounds__(256)
void stats_partial(const float* __restrict__ Y, float* __restrict__ part) {
    const int c = blockIdx.y, blk = blockIdx.x, t = threadIdx.x;
    const float* p = Y + (size_t)c * M_ + (size_t)blk * 8192;
    float s = 0.0f, s2 = 0.0f;
#pragma unroll 4
    for (int i = 0; i < 32; ++i) {
        float v = p[t + i * 256];
        s += v; s2 += v * v;
    }
    __shared__ float ls[256], lq[256];
    ls[t] = s; lq[t] = s2;
    __syncthreads();
    for (int off = 128; off >= 1; off >>= 1) {
        if (t < off) { ls[t] += ls[t + off]; lq[t] += lq[t + off]; }
        __syncthreads();
    }
    if (t == 0) {
        part[(c * 128 + blk) * 2 + 0] = ls[0];
        part[(c * 128 + blk) * 2 + 1] = lq[0];
    }
}

__global__ __launch_bounds__(128)
void stats_reduce(const float* __restrict__ part, float* __restrict__ stats) {
    const int c = blockIdx.x, t = threadIdx.x;
    __shared__ float ls[128], lq[128];
    ls[t] = part[(c * 128 + t) * 2 + 0];
    lq[t] = part[(c * 128 + t) * 2 + 1];
    __syncthreads();
    for (int off = 64; off >= 1; off >>= 1) {
        if (t < off) { ls[t] += ls[t + off]; lq[t] += lq[t + off]; }
        __syncthreads();
    }
    if (t == 0) { stats[c * 2 + 0] = ls[0]; stats[c * 2 + 1] = lq[0]; }
}

// ------------------------------------------------ BN (train) + ReLU --------
__global__ __launch_bounds__(256)
void bnrelu(float* __restrict__ Y, const float* __restrict__ stats,
            const float* __restrict__ g, const float* __restrict__ bb) {
    size_t i = (size_t)blockIdx.x * 256 + threadIdx.x;
    int c = (int)(i >> 20);                      // M_ == 2^20
    float mean = stats[c * 2 + 0] * (1.0f / M_);
    float var  = stats[c * 2 + 1] * (1.0f / M_) - mean * mean;
    float sc = g[c] * rsqrtf(var + EPSBN);
    float sh = bb[c] - mean * sc;
    float v = Y[i] * sc + sh;
    Y[i] = v > 0.0f ? v : 0.0f;
}

// ------------------------------- BN3 + ReLU + max over nsample -> output ---
__global__ __launch_bounds__(256)
void maxpool_bn(const float* __restrict__ Y3, const float* __restrict__ stats,
                const float* __restrict__ g, const float* __restrict__ bb,
                float* __restrict__ out) {
    int i = blockIdx.x * 256 + threadIdx.x;      // c * 16384 + group
    int c = i >> 14;
    int grp = i & 16383;
    int b = grp >> 11, s = grp & 2047;
    float mean = stats[c * 2 + 0] * (1.0f / M_);
    float var  = stats[c * 2 + 1] * (1.0f / M_) - mean * mean;
    float sc = g[c] * rsqrtf(var + EPSBN);
    float sh = bb[c] - mean * sc;
    const float* p = Y3 + (size_t)c * M_ + (size_t)grp * NS;
    float mx = -1e30f;
#pragma unroll 4
    for (int j = 0; j < NS; ++j) {
        float v = p[j] * sc + sh;
        v = v > 0.0f ? v : 0.0f;
        mx = fmaxf(mx, v);
    }
    out[(size_t)b * 64 * S_ + (size_t)c * S_ + s] = mx;
}

// ---------------------------------------------------------------------------
extern "C" void kernel_launch(void* const* d_in, const int* in_sizes, int n_in,
                              void* d_out, int out_size, void* d_ws, size_t ws_size,
                              hipStream_t stream) {
    const float* xyz   = (const float*)d_in[0];   // (B,N,3)
    const float* feats = (const float*)d_in[1];   // (B,16,N)
    const float* W1    = (const float*)d_in[2];   // (32,19)
    const float* g1    = (const float*)d_in[3];
    const float* b1    = (const float*)d_in[4];
    const float* W2    = (const float*)d_in[5];   // (32,32)
    const float* g2    = (const float*)d_in[6];
    const float* b2    = (const float*)d_in[7];
    const float* W3    = (const float*)d_in[8];   // (64,32)
    const float* g3    = (const float*)d_in[9];
    const float* b3    = (const float*)d_in[10];

    float* out_xyz  = (float*)d_out;                       // (B,S,3)
    float* out_feat = (float*)d_out + (size_t)B_ * S_ * 3; // (B,64,S)

    char* ws = (char*)d_ws;
    int*   fps_idx = (int*)ws;                        // 64 KB
    float* part    = (float*)(ws + (64 << 10));       // 64 KB partials
    float* stats   = (float*)(ws + (128 << 10));      // per-layer stats
    float* Xb      = (float*)(ws + (1 << 18));        // 20 x M  (80 MB)
    float* Y1      = Xb + (size_t)20 * M_;            // 32 x M  (128 MB)
    float* Y2      = Y1 + (size_t)32 * M_;            // 32 x M  (128 MB)
    float* Y3      = Y2 + (size_t)32 * M_;            // 64 x M  (256 MB)

    // 1) FPS + new_xyz
    fps_kernel<<<B_, 1024, 0, stream>>>(xyz, fps_idx, out_xyz);

    // 2) Ball query + grouped matrix X (19 real rows + zero pad row)
    ball_group_kernel<<<2048, 256, 0, stream>>>(xyz, feats, out_xyz, Xb);

    // Each GEMM wave covers 16 x 64 output; grid.x = M / (8 waves * 64) = 2048
    // 3) Layer 1: Y1 = W1 @ X   (K = 19 padded to 20)
    gemm_wmma<19, 20, 4><<<dim3(2048, 2), 256, 0, stream>>>(W1, Xb, Y1);
    stats_partial<<<dim3(128, 32), 256, 0, stream>>>(Y1, part);
    stats_reduce<<<32, 128, 0, stream>>>(part, stats);
    bnrelu<<<(32u * M_) / 256, 256, 0, stream>>>(Y1, stats, g1, b1);

    // 4) Layer 2: Y2 = W2 @ Y1n
    gemm_wmma<32, 32, 4><<<dim3(2048, 2), 256, 0, stream>>>(W2, Y1, Y2);
    stats_partial<<<dim3(128, 32), 256, 0, stream>>>(Y2, part);
    stats_reduce<<<32, 128, 0, stream>>>(part, stats);
    bnrelu<<<(32u * M_) / 256, 256, 0, stream>>>(Y2, stats, g2, b2);

    // 5) Layer 3: Y3 = W3 @ Y2n, then BN3+ReLU+max fused into pooling
    gemm_wmma<32, 32, 4><<<dim3(2048, 4), 256, 0, stream>>>(W3, Y2, Y3);
    stats_partial<<<dim3(128, 64), 256, 0, stream>>>(Y3, part);
    stats_reduce<<<64, 128, 0, stream>>>(part, stats);
    maxpool_bn<<<(64 * B_ * S_) / 256, 256, 0, stream>>>(Y3, stats, g3, b3, out_feat);
}